// DifferentiableRasterizer_56796647522831
// MI455X (gfx1250) — compile-verified
//
#include <hip/hip_runtime.h>
#include <hip/hip_bf16.h>
#include <math.h>

typedef float v2f __attribute__((ext_vector_type(2)));
typedef float v8f __attribute__((ext_vector_type(8)));

#define NB 4
#define NPTS 8192
#define NTRI 1024
#define TPB 256            // 8 waves (wave32)
#define PTS_PER_WAVE 16
#define PTS_PER_BLOCK 128  // 8 waves * 16 points
#define TRI_CHUNK 256
#define TAB_STRIDE 36      // floats per triangle table row (bank-conflict friendly, 8B aligned)

#if defined(__has_builtin)
#if __has_builtin(__builtin_amdgcn_global_load_async_to_lds_b32)
#define HAVE_ASYNC_LDS 1
#endif
#endif

typedef __attribute__((address_space(1))) int GInt;   // global (AS1) int
typedef __attribute__((address_space(3))) int LInt;   // LDS (AS3) int

__global__ __launch_bounds__(TPB) void raster_wmma_kernel(
    const float* __restrict__ points,   // [NB][NPTS][3]
    const float* __restrict__ tris,     // [NB][NTRI][3][3]
    float* __restrict__ out)            // [NB][NPTS]
{
    __shared__ __align__(16) float tab[TRI_CHUNK * TAB_STRIDE]; // per-tri precomputed table
    __shared__ __align__(16) float raw[TRI_CHUNK * 9];          // staged raw triangle verts

    const int tid   = threadIdx.x;
    const int lane  = tid & 31;
    const int wid   = tid >> 5;
    const int batch = blockIdx.y;
    const int tileBase = blockIdx.x * PTS_PER_BLOCK + wid * PTS_PER_WAVE;

    // ---- A matrix: 16x4 f32 (points, K padded with 1.0) ----
    // lanes 0-15: VGPR0=K0(px), VGPR1=K1(py); lanes 16-31: VGPR0=K2(pz), VGPR1=K3(1.0)
    const float* P = points + (size_t)batch * NPTS * 3;
    const int pidx = tileBase + (lane & 15);
    const float px = P[pidx * 3 + 0];
    const float py = P[pidx * 3 + 1];
    const float pz = P[pidx * 3 + 2];
    v2f A;
    if (lane < 16) { A.x = px; A.y = py; } else { A.x = pz; A.y = 1.0f; }

    // |p|^2 for each of the wave's 16 points, arranged per C/D slot layout:
    // slot m holds point (m + 8*(lane>=16)) of the tile.
    const float ppv = px * px + py * py + pz * pz;
    float ppm[8];
#pragma unroll
    for (int m = 0; m < 8; ++m)
        ppm[m] = __shfl(ppv, m + ((lane & 16) >> 1), 32);

    float mind[8];
#pragma unroll
    for (int m = 0; m < 8; ++m) mind[m] = 3.0e38f;

    const float* Tg = tris + (size_t)batch * NTRI * 9;

    for (int c0 = 0; c0 < NTRI; c0 += TRI_CHUNK) {
        __syncthreads();  // protect LDS reuse from previous chunk

        // ---- stage 256 triangles (9 floats each) global -> LDS ----
#if defined(HAVE_ASYNC_LDS)
        {
            GInt* g = (GInt*)(Tg + (size_t)(c0 + tid) * 9);
            LInt* l = (LInt*)&raw[tid * 9];
            __builtin_amdgcn_global_load_async_to_lds_b32(g, l,  0, 0);
            __builtin_amdgcn_global_load_async_to_lds_b32(g, l,  4, 0);
            __builtin_amdgcn_global_load_async_to_lds_b32(g, l,  8, 0);
            __builtin_amdgcn_global_load_async_to_lds_b32(g, l, 12, 0);
            __builtin_amdgcn_global_load_async_to_lds_b32(g, l, 16, 0);
            __builtin_amdgcn_global_load_async_to_lds_b32(g, l, 20, 0);
            __builtin_amdgcn_global_load_async_to_lds_b32(g, l, 24, 0);
            __builtin_amdgcn_global_load_async_to_lds_b32(g, l, 28, 0);
            __builtin_amdgcn_global_load_async_to_lds_b32(g, l, 32, 0);
            asm volatile("s_wait_asynccnt 0x0" ::: "memory");
        }
#else
        {
#pragma unroll
            for (int k = 0; k < 9; ++k)
                raw[tid * 9 + k] = Tg[(size_t)(c0 + tid) * 9 + k];
        }
#endif
        __syncthreads();

        // ---- per-triangle invariant precompute (1 tri / thread) ----
        {
            const float* r = &raw[tid * 9];
            const float v0x = r[0], v0y = r[1], v0z = r[2];
            const float v1x = r[3], v1y = r[4], v1z = r[5];
            const float v2x = r[6], v2y = r[7], v2z = r[8];
            const float e0x = v1x - v0x, e0y = v1y - v0y, e0z = v1z - v0z;
            const float e1x = v2x - v0x, e1y = v2y - v0y, e1z = v2z - v0z;
            const float e2x = v2x - v1x, e2y = v2y - v1y, e2z = v2z - v1z;
            const float nx = e0y * e1z - e0z * e1y;
            const float ny = e0z * e1x - e0x * e1z;
            const float nz = e0x * e1y - e0y * e1x;
            const float dot00 = e0x * e0x + e0y * e0y + e0z * e0z;
            const float dot01 = e0x * e1x + e0y * e1y + e0z * e1z;
            const float dot11 = e1x * e1x + e1y * e1y + e1z * e1z;
            const float dot22 = e2x * e2x + e2y * e2y + e2z * e2z;
            const float area2 = nx * nx + ny * ny + nz * nz;
            const float invden = 1.0f / fmaxf(area2, 1e-12f);
            const float inv00 = 1.0f / fmaxf(dot00, 1e-12f);
            const float inv11 = 1.0f / fmaxf(dot11, 1e-12f);
            const float inv22 = 1.0f / fmaxf(dot22, 1e-12f);
            const float b0 = v0x * e0x + v0y * e0y + v0z * e0z;
            const float b1 = v0x * e1x + v0y * e1y + v0z * e1z;
            const float b2 = v1x * e2x + v1y * e2y + v1z * e2z;
            const float bn = v0x * nx + v0y * ny + v0z * nz;
            const float vv0 = v0x * v0x + v0y * v0y + v0z * v0z;
            const float aok = (area2 >= 4e-10f) ? 1.0f : 0.0f;
            float* w = &tab[tid * TAB_STRIDE];
            w[0] = e0x;  w[1] = e0y;  w[2] = e0z;  w[3] = -b0;   // f0: e0.d
            w[4] = e1x;  w[5] = e1y;  w[6] = e1z;  w[7] = -b1;   // f1: e1.d
            w[8] = e2x;  w[9] = e2y;  w[10] = e2z; w[11] = -b2;  // f2: e2.(p-v1)
            w[12] = nx;  w[13] = ny;  w[14] = nz;  w[15] = -bn;  // fn: n.d
            w[16] = -2.0f * v0x; w[17] = -2.0f * v0y; w[18] = -2.0f * v0z; w[19] = vv0; // fd
            w[20] = dot00; w[21] = dot01; w[22] = dot11; w[23] = dot22;
            w[24] = invden; w[25] = inv00; w[26] = inv11; w[27] = inv22;
            w[28] = aok;
        }
        __syncthreads();

        // ---- 16 triangle tiles of 16; 5 WMMAs per tile ----
        const int so = (lane & 16) >> 3;  // 0 for K0/K1 half, 2 for K2/K3 half
        for (int nt = 0; nt < TRI_CHUNK / 16; ++nt) {
            const float* T = &tab[(nt * 16 + (lane & 15)) * TAB_STRIDE];
            const v2f B0 = *(const v2f*)&T[0 + so];
            const v2f B1 = *(const v2f*)&T[4 + so];
            const v2f B2 = *(const v2f*)&T[8 + so];
            const v2f Bn = *(const v2f*)&T[12 + so];
            const v2f Bd = *(const v2f*)&T[16 + so];
            const float dot00 = T[20], dot01 = T[21], dot11 = T[22], dot22 = T[23];
            const float invden = T[24], inv00 = T[25], inv11 = T[26], inv22 = T[27];
            const float aok = T[28];

            v8f z = {};
            v8f W0 = __builtin_amdgcn_wmma_f32_16x16x4_f32(false, A, false, B0, (short)0, z, false, false);
            v8f W1 = __builtin_amdgcn_wmma_f32_16x16x4_f32(false, A, false, B1, (short)0, z, false, false);
            v8f W2 = __builtin_amdgcn_wmma_f32_16x16x4_f32(false, A, false, B2, (short)0, z, false, false);
            v8f Wn = __builtin_amdgcn_wmma_f32_16x16x4_f32(false, A, false, Bn, (short)0, z, false, false);
            v8f Wd = __builtin_amdgcn_wmma_f32_16x16x4_f32(false, A, false, Bd, (short)0, z, false, false);

#pragma unroll
            for (int m = 0; m < 8; ++m) {
                const float w0 = W0[m];
                const float w1 = W1[m];
                const float w2 = W2[m];
                const float wn = Wn[m];
                const float dd = Wd[m] + ppm[m];          // |p - v0|^2
                const float u = (dot11 * w0 - dot01 * w1) * invden;
                const float v = (dot00 * w1 - dot01 * w0) * invden;
                const float plane = wn * wn * invden;
                const bool inside = (u >= 0.0f) & (v >= 0.0f) & ((u + v) <= 1.0f) & (aok > 0.5f);
                const float t0 = fminf(fmaxf(w0 * inv00, 0.0f), 1.0f);
                const float d0 = dd + t0 * (t0 * dot00 - 2.0f * w0);
                const float t1 = fminf(fmaxf(w1 * inv11, 0.0f), 1.0f);
                const float d1 = dd + t1 * (t1 * dot11 - 2.0f * w1);
                const float dd1 = dd - 2.0f * w0 + dot00; // |p - v1|^2
                const float t2 = fminf(fmaxf(w2 * inv22, 0.0f), 1.0f);
                const float d2 = dd1 + t2 * (t2 * dot22 - 2.0f * w2);
                const float edge = fminf(d0, fminf(d1, d2));
                const float dsq = inside ? plane : edge;
                mind[m] = fminf(mind[m], dsq);
            }
        }
    }

    // ---- per-point min across the 16 N-lanes of each half-wave, then store ----
    float* O = out + (size_t)batch * NPTS;
#pragma unroll
    for (int m = 0; m < 8; ++m) {
        float v = mind[m];
        v = fminf(v, __shfl_xor(v, 1, 32));
        v = fminf(v, __shfl_xor(v, 2, 32));
        v = fminf(v, __shfl_xor(v, 4, 32));
        v = fminf(v, __shfl_xor(v, 8, 32));
        if ((lane & 15) == m) {
            const int pt = tileBase + m + ((lane & 16) >> 1); // lower half: m, upper half: m+8
            O[pt] = __expf(-100.0f * v);
        }
    }
}

extern "C" void kernel_launch(void* const* d_in, const int* in_sizes, int n_in,
                              void* d_out, int out_size, void* d_ws, size_t ws_size,
                              hipStream_t stream) {
    const float* points = (const float*)d_in[0];
    const float* tris   = (const float*)d_in[1];
    float* out          = (float*)d_out;
    dim3 grid(NPTS / PTS_PER_BLOCK, NB);
    raster_wmma_kernel<<<grid, dim3(TPB), 0, stream>>>(points, tris, out);
}